// SparseGraphLearn_85718957293637
// MI455X (gfx1250) — compile-verified
//
#include <hip/hip_runtime.h>
#include <hip/hip_bf16.h>

typedef __attribute__((ext_vector_type(2))) float v2f;
typedef __attribute__((ext_vector_type(8))) float v8f;

#define D_IN 512
#define D_OUT 64

// ---------------------------------------------------------------------------
// Kernel 0: transpose W [512,64] -> Wt [64,512] so B fragments are contiguous
// ---------------------------------------------------------------------------
__global__ void wt_transpose_kernel(const float* __restrict__ w,
                                    float* __restrict__ wt) {
    int idx = blockIdx.x * blockDim.x + threadIdx.x;
    if (idx < D_IN * D_OUT) {
        int r = idx / D_OUT;   // k
        int c = idx % D_OUT;   // n
        wt[c * D_IN + r] = w[idx];
    }
}

// ---------------------------------------------------------------------------
// Kernel 1: zero row-max bits and row sums (workspace is poisoned by harness)
// ---------------------------------------------------------------------------
__global__ void zero_kernel(float* __restrict__ p, int n) {
    int idx = blockIdx.x * blockDim.x + threadIdx.x;
    if (idx < n) p[idx] = 0.0f;
}

// ---------------------------------------------------------------------------
// Kernel 2: h = x @ W via V_WMMA_F32_16X16X4_F32.
// One wave -> one 16x64 output tile (4 accumulators), K looped in steps of 4.
// A (16x4 f32): lanes 0-15 K={0,1}, lanes 16-31 K={2,3}   (ISA 7.12.2)
// B (4x16 f32): lanes 0-15 N=lane K={0,1}, lanes 16-31 N=lane-16 K={2,3}
// C/D (16x16):  VGPR r -> M=r (lanes 0-15) / M=8+r (lanes 16-31), N=lane&15
// Tile guard is wave-uniform so EXEC is all-1s at every WMMA.
// ---------------------------------------------------------------------------
__global__ __launch_bounds__(128) void gemm_wmma_kernel(
    const float* __restrict__ x, const float* __restrict__ wt,
    float* __restrict__ h, int nTiles) {
    int lane = threadIdx.x & 31;
    int wave = threadIdx.x >> 5;
    int rowTile = blockIdx.x * 4 + wave;
    if (rowTile >= nTiles) return;

    int sub  = lane & 15;
    int kgrp = lane >> 4;  // 0: lanes 0-15, 1: lanes 16-31

    const float* xrow = x + (size_t)(rowTile * 16 + sub) * D_IN + 2 * kgrp;
    const float* wt0 = wt + (size_t)(0  + sub) * D_IN + 2 * kgrp;
    const float* wt1 = wt + (size_t)(16 + sub) * D_IN + 2 * kgrp;
    const float* wt2 = wt + (size_t)(32 + sub) * D_IN + 2 * kgrp;
    const float* wt3 = wt + (size_t)(48 + sub) * D_IN + 2 * kgrp;

    v8f c0 = {}; v8f c1 = {}; v8f c2 = {}; v8f c3 = {};

#pragma unroll 4
    for (int k = 0; k < D_IN; k += 4) {
        v2f a  = *(const v2f*)(xrow + k);
        v2f b0 = *(const v2f*)(wt0 + k);
        v2f b1 = *(const v2f*)(wt1 + k);
        v2f b2 = *(const v2f*)(wt2 + k);
        v2f b3 = *(const v2f*)(wt3 + k);
        c0 = __builtin_amdgcn_wmma_f32_16x16x4_f32(false, a, false, b0,
                                                   (short)0, c0, false, false);
        c1 = __builtin_amdgcn_wmma_f32_16x16x4_f32(false, a, false, b1,
                                                   (short)0, c1, false, false);
        c2 = __builtin_amdgcn_wmma_f32_16x16x4_f32(false, a, false, b2,
                                                   (short)0, c2, false, false);
        c3 = __builtin_amdgcn_wmma_f32_16x16x4_f32(false, a, false, b3,
                                                   (short)0, c3, false, false);
    }

    int rowBase = rowTile * 16 + kgrp * 8;
    float* hp = h + (size_t)rowBase * D_OUT + sub;
#pragma unroll
    for (int r = 0; r < 8; ++r) {
        hp[(size_t)r * D_OUT + 0]  = c0[r];
        hp[(size_t)r * D_OUT + 16] = c1[r];
        hp[(size_t)r * D_OUT + 32] = c2[r];
        hp[(size_t)r * D_OUT + 48] = c3[r];
    }
}

// ---------------------------------------------------------------------------
// Kernel 3: per-edge score s = relu(|h_src - h_dst| . a); atomic row max.
// 16 lanes per edge: each lane loads a float4 slice (coalesced 256B row read,
// h is L2-resident). Butterfly shfl reduction within the 16-lane group.
// Scores are >= 0 so float max == uint-bits max (init 0 == 0.0f, matching the
// reference's isinf->0 substitution for empty rows).
// ---------------------------------------------------------------------------
__global__ void edge_score_kernel(const float* __restrict__ h,
                                  const long long* __restrict__ edge,
                                  const float* __restrict__ a,
                                  float* __restrict__ edge_v,
                                  unsigned int* __restrict__ row_max_bits,
                                  int E) {
    int lane = threadIdx.x & 31;
    int waveInBlock = threadIdx.x >> 5;
    int wavesPerBlock = blockDim.x >> 5;
    long long waveId = (long long)blockIdx.x * wavesPerBlock + waveInBlock;
    long long e = waveId * 2 + (lane >> 4);
    int sub = lane & 15;
    if (e < E) {
        int src = (int)edge[e];
        int dst = (int)edge[(long long)E + e];
        const float4* hs = (const float4*)(h + (size_t)src * D_OUT);
        const float4* hd = (const float4*)(h + (size_t)dst * D_OUT);
        float4 s4 = hs[sub];
        float4 d4 = hd[sub];
        float4 a4 = ((const float4*)a)[sub];
        float p = fabsf(s4.x - d4.x) * a4.x + fabsf(s4.y - d4.y) * a4.y +
                  fabsf(s4.z - d4.z) * a4.z + fabsf(s4.w - d4.w) * a4.w;
        // reduce across the 16-lane subgroup (xor<16 stays inside the group)
        p += __shfl_xor(p, 1, 32);
        p += __shfl_xor(p, 2, 32);
        p += __shfl_xor(p, 4, 32);
        p += __shfl_xor(p, 8, 32);
        if (sub == 0) {
            float s = p > 0.0f ? p : 0.0f;
            edge_v[e] = s;
            atomicMax(&row_max_bits[src], __float_as_uint(s));
        }
    }
}

// ---------------------------------------------------------------------------
// Kernel 4: exp_v = exp(s - max[row]); atomic row sum. Overwrites edge_v.
// ---------------------------------------------------------------------------
__global__ void edge_exp_kernel(const long long* __restrict__ edge,
                                const float* __restrict__ row_max,
                                float* __restrict__ edge_v,
                                float* __restrict__ row_sum, int E) {
    int idx = blockIdx.x * blockDim.x + threadIdx.x;
    if (idx < E) {
        int src = (int)edge[idx];
        float ev = __expf(edge_v[idx] - row_max[src]);
        edge_v[idx] = ev;
        atomicAdd(&row_sum[src], ev);
    }
}

// ---------------------------------------------------------------------------
// Kernel 5: adj = exp_v / (sum[row] + 1e-16)
// ---------------------------------------------------------------------------
__global__ void edge_norm_kernel(const long long* __restrict__ edge,
                                 const float* __restrict__ edge_v,
                                 const float* __restrict__ row_sum,
                                 float* __restrict__ adj, int E) {
    int idx = blockIdx.x * blockDim.x + threadIdx.x;
    if (idx < E) {
        int src = (int)edge[idx];
        adj[idx] = edge_v[idx] / (row_sum[src] + 1e-16f);
    }
}

extern "C" void kernel_launch(void* const* d_in, const int* in_sizes, int n_in,
                              void* d_out, int out_size, void* d_ws, size_t ws_size,
                              hipStream_t stream) {
    const float*     x    = (const float*)d_in[0];
    const long long* edge = (const long long*)d_in[1];  // int64 [2, E]
    // d_in[2] = num_nodes scalar (derived from in_sizes instead)
    const float*     w    = (const float*)d_in[3];      // [512, 64]
    const float*     a    = (const float*)d_in[4];      // [64, 1]

    const int N = in_sizes[0] / D_IN;   // 100000
    const int E = in_sizes[1] / 2;      // 1600000

    float* h   = (float*)d_out;                       // [N, 64]
    float* adj = (float*)d_out + (size_t)N * D_OUT;   // [E]

    // Workspace layout (floats): [0,N) row-max bits | [N,2N) row sums |
    // [2N,2N+E) per-edge scratch | [2N+E, +32768) W transposed.
    float* ws_f            = (float*)d_ws;
    unsigned int* max_bits = (unsigned int*)ws_f;
    float* row_max         = ws_f;             // same memory, read as float
    float* row_sum         = ws_f + N;
    float* edge_v          = ws_f + 2 * (size_t)N;
    float* wt              = ws_f + 2 * (size_t)N + E;

    // 0) transpose W
    wt_transpose_kernel<<<(D_IN * D_OUT + 255) / 256, 256, 0, stream>>>(w, wt);
    // 1) zero max + sum (contiguous 2N floats)
    zero_kernel<<<(2 * N + 255) / 256, 256, 0, stream>>>(ws_f, 2 * N);
    // 2) GEMM via fp32 WMMA: one wave per 16x64 tile, 4 waves per block
    int nTiles = (N + 15) / 16;
    gemm_wmma_kernel<<<(nTiles + 3) / 4, 128, 0, stream>>>(x, wt, h, nTiles);
    // 3) edge scores + segment max (16 edges per 256-thread block)
    edge_score_kernel<<<(E + 15) / 16, 256, 0, stream>>>(h, edge, a, edge_v,
                                                         max_bits, E);
    // 4) exp + segment sum
    edge_exp_kernel<<<(E + 255) / 256, 256, 0, stream>>>(edge, row_max, edge_v,
                                                         row_sum, E);
    // 5) normalize
    edge_norm_kernel<<<(E + 255) / 256, 256, 0, stream>>>(edge, edge_v, row_sum,
                                                          adj, E);
}